// GCN_76201309766165
// MI455X (gfx1250) — compile-verified
//
#include <hip/hip_runtime.h>
#include <hip/hip_bf16.h>

#define D_FEAT 64
#define UNROLL 4

// ---------------------------------------------------------------------------
// Pass 1: per-edge scatter. One wave32 per edge; 4 edges in flight per wave.
// Lane l covers features [2l, 2l+1]: each edge is one coalesced b64 gather of
// the source row (256 B per wave) + two global_atomic_add_f32 (no-return,
// executed at L2) into the destination row. in_feat (25.6 MB) and the
// accumulator (25.6 MB) both stay resident in the 192 MB L2, so the random
// gathers and all 109M atomics resolve at L2 bandwidth; HBM only sees the
// compulsory ~130 MB. Edge indices are wave-uniform -> readfirstlane puts
// them in SGPRs so addresses use the scalar-base + lane-offset form.
// ---------------------------------------------------------------------------
__global__ void __launch_bounds__(256) gcn_edge_scatter(
    const float* __restrict__ in_feat,
    const int*   __restrict__ src,
    const int*   __restrict__ dst,
    float*       __restrict__ agg,   // == d_out, zero-initialized
    float*       __restrict__ deg,   // in d_ws, zero-initialized
    int n_edges)
{
    const int lane   = threadIdx.x & 31;
    const int wib    = threadIdx.x >> 5;
    const int wpb    = blockDim.x >> 5;
    const int waveId = blockIdx.x * wpb + wib;
    const int nWaves = gridDim.x * wpb;

    const int lo = 2 * lane;             // this lane's feature offset
    int e = waveId;

    // Main loop: 4 independent gathers in flight overlap with the atomic
    // drain of the previous group (atomics are fire-and-forget).
    for (; e + 3 * nWaves < n_edges; e += UNROLL * nWaves) {
        const int s0 = __builtin_amdgcn_readfirstlane(src[e + 0 * nWaves]);
        const int s1 = __builtin_amdgcn_readfirstlane(src[e + 1 * nWaves]);
        const int s2 = __builtin_amdgcn_readfirstlane(src[e + 2 * nWaves]);
        const int s3 = __builtin_amdgcn_readfirstlane(src[e + 3 * nWaves]);
        const int d0 = __builtin_amdgcn_readfirstlane(dst[e + 0 * nWaves]);
        const int d1 = __builtin_amdgcn_readfirstlane(dst[e + 1 * nWaves]);
        const int d2 = __builtin_amdgcn_readfirstlane(dst[e + 2 * nWaves]);
        const int d3 = __builtin_amdgcn_readfirstlane(dst[e + 3 * nWaves]);

        // Prefetch the next group's source rows (gfx1250 global_prefetch_b8).
        const int en = e + UNROLL * nWaves;
        if (en + 3 * nWaves < n_edges) {
#pragma unroll
            for (int k = 0; k < UNROLL; ++k) {
                const int sp = src[en + k * nWaves];
                __builtin_prefetch(in_feat + (size_t)sp * D_FEAT + lo, 0, 3);
            }
        }

        const float2 v0 = *(const float2*)(in_feat + (size_t)s0 * D_FEAT + lo);
        const float2 v1 = *(const float2*)(in_feat + (size_t)s1 * D_FEAT + lo);
        const float2 v2 = *(const float2*)(in_feat + (size_t)s2 * D_FEAT + lo);
        const float2 v3 = *(const float2*)(in_feat + (size_t)s3 * D_FEAT + lo);

        float* p0 = agg + (size_t)d0 * D_FEAT + lo;
        float* p1 = agg + (size_t)d1 * D_FEAT + lo;
        float* p2 = agg + (size_t)d2 * D_FEAT + lo;
        float* p3 = agg + (size_t)d3 * D_FEAT + lo;

        unsafeAtomicAdd(p0,     v0.x);   // -> global_atomic_add_f32
        unsafeAtomicAdd(p0 + 1, v0.y);
        unsafeAtomicAdd(p1,     v1.x);
        unsafeAtomicAdd(p1 + 1, v1.y);
        unsafeAtomicAdd(p2,     v2.x);
        unsafeAtomicAdd(p2 + 1, v2.y);
        unsafeAtomicAdd(p3,     v3.x);
        unsafeAtomicAdd(p3 + 1, v3.y);

        if (lane == 0) {                 // degree counts, exact << 2^24
            unsafeAtomicAdd(deg + d0, 1.0f);
            unsafeAtomicAdd(deg + d1, 1.0f);
            unsafeAtomicAdd(deg + d2, 1.0f);
            unsafeAtomicAdd(deg + d3, 1.0f);
        }
    }

    // Tail: remaining edges, one at a time.
    for (; e < n_edges; e += nWaves) {
        const int s = __builtin_amdgcn_readfirstlane(src[e]);
        const int d = __builtin_amdgcn_readfirstlane(dst[e]);
        const float2 v = *(const float2*)(in_feat + (size_t)s * D_FEAT + lo);
        float* p = agg + (size_t)d * D_FEAT + lo;
        unsafeAtomicAdd(p,     v.x);
        unsafeAtomicAdd(p + 1, v.y);
        if (lane == 0) unsafeAtomicAdd(deg + d, 1.0f);
    }
}

// ---------------------------------------------------------------------------
// Pass 2: out = agg / deg + in_feat. One float4 per thread, fully coalesced
// b128 traffic. deg >= 1 always (self loops appended by the reference).
// ---------------------------------------------------------------------------
__global__ void __launch_bounds__(256) gcn_finalize(
    const float* __restrict__ in_feat,
    const float* __restrict__ deg,
    float*       __restrict__ out,
    int n_vec4)   // n_nodes * 16
{
    const int tid = blockIdx.x * blockDim.x + threadIdx.x;
    if (tid >= n_vec4) return;

    const int node = tid >> 4;             // 16 float4 per node row
    const float inv = 1.0f / deg[node];    // precise div, once per thread

    const float4 a = ((const float4*)out)[tid];
    const float4 f = ((const float4*)in_feat)[tid];
    float4 o;
    o.x = a.x * inv + f.x;
    o.y = a.y * inv + f.y;
    o.z = a.z * inv + f.z;
    o.w = a.w * inv + f.w;
    ((float4*)out)[tid] = o;
}

extern "C" void kernel_launch(void* const* d_in, const int* in_sizes, int n_in,
                              void* d_out, int out_size, void* d_ws, size_t ws_size,
                              hipStream_t stream) {
    const float* in_feat = (const float*)d_in[0];
    const int*   src     = (const int*)d_in[1];
    const int*   dst     = (const int*)d_in[2];

    const int n_feat_elems = in_sizes[0];            // n_nodes * 64
    const int n_edges      = in_sizes[1];
    const int n_nodes      = n_feat_elems / D_FEAT;

    float* out = (float*)d_out;
    float* deg = (float*)d_ws;                       // n_nodes floats of scratch

    // Zero the accumulator (d_out doubles as agg) and the degree array.
    hipMemsetAsync(out, 0, (size_t)n_feat_elems * sizeof(float), stream);
    hipMemsetAsync(deg, 0, (size_t)n_nodes * sizeof(float), stream);

    // Pass 1: 4096 blocks x 8 waves = 32768 waves, ~52 edges per wave,
    // processed 4 at a time.
    const int blocks_edge = 4096;
    gcn_edge_scatter<<<blocks_edge, 256, 0, stream>>>(
        in_feat, src, dst, out, deg, n_edges);

    // Pass 2: one float4 per thread.
    const int n_vec4 = n_nodes * (D_FEAT / 4);
    const int blocks_fin = (n_vec4 + 255) / 256;
    gcn_finalize<<<blocks_fin, 256, 0, stream>>>(in_feat, deg, out, n_vec4);
}